// GCN_49005576847377
// MI455X (gfx1250) — compile-verified
//
#include <hip/hip_runtime.h>

typedef __attribute__((ext_vector_type(2))) float v2f;
typedef __attribute__((ext_vector_type(8))) float v8f;

#define D 64
#define TILE 16

// ---------------- zero workspace (agg + deg) ----------------
__global__ void GCN_zero_ws(float* __restrict__ p, size_t n) {
  size_t i = (size_t)blockIdx.x * blockDim.x + threadIdx.x;
  size_t stride = (size_t)gridDim.x * blockDim.x;
  for (; i < n; i += stride) p[i] = 0.0f;
}

// ---------------- edge scatter: agg[dst] += h[src]; deg[src] += 1 ----------------
// 16 threads per edge; each thread moves a float4 (4 dims) with 4 f32 atomics.
// h (12.8MB) and agg (12.8MB) are L2-resident on MI455X (192MB L2).
__global__ void GCN_scatter_deg(const float* __restrict__ h,
                                const int* __restrict__ src,
                                const int* __restrict__ dst,
                                float* __restrict__ agg,
                                float* __restrict__ deg,
                                int n_edges) {
  int t = blockIdx.x * blockDim.x + threadIdx.x;
  int e = t >> 4;
  int part = t & 15;
  if (e >= n_edges) return;
  int s = src[e];
  int d = dst[e];
  const float4* hp = (const float4*)(h + (size_t)s * D);
  float4 hv = hp[part];
  float* ap = agg + (size_t)d * D + part * 4;
  atomicAdd(ap + 0, hv.x);
  atomicAdd(ap + 1, hv.y);
  atomicAdd(ap + 2, hv.z);
  atomicAdd(ap + 3, hv.w);
  if (part == 0) atomicAdd(deg + s, 1.0f);
}

// ---------------- fused GEMM + norm + bias + relu via V_WMMA_F32_16X16X4_F32 ----------------
// One wave32 per 16x16 output tile. kernel matrix staged in LDS (16KB).
// A (16x4 f32): lanes 0-15 -> M=lane, v0=K+0, v1=K+1 ; lanes 16-31 -> M=lane-16, v0=K+2, v1=K+3
// B (4x16 f32): lanes 0-15 -> N=lane, v0=K+0, v1=K+1 ; lanes 16-31 -> N=lane-16, v0=K+2, v1=K+3
// C/D (16x16 f32, 8 VGPRs): VGPR r, lanes 0-15 -> (M=r, N=lane); lanes 16-31 -> (M=r+8, N=lane-16)
__global__ void __launch_bounds__(256)
GCN_gemm_norm_bias_relu(const float* __restrict__ agg,
                        const float* __restrict__ kern,
                        const float* __restrict__ bias,
                        const float* __restrict__ deg,
                        float* __restrict__ out,
                        int n_tiles_m) {
  __shared__ float kbuf[D * D];
  for (int i = threadIdx.x; i < D * D; i += 256) kbuf[i] = kern[i];
  __syncthreads();

  int wave = threadIdx.x >> 5;   // wave32: 8 waves per 256-thread block
  int lane = threadIdx.x & 31;
  int tid = blockIdx.x * 8 + wave;
  int total_tiles = n_tiles_m * (D / TILE);
  if (tid >= total_tiles) return;      // wave-uniform exit: EXEC stays all-ones for WMMA

  int row_tile = tid >> 2;             // D/TILE == 4 col tiles
  int col_tile = tid & 3;
  int m0 = row_tile * TILE;
  int n0 = col_tile * TILE;

  int half = lane >> 4;                // 0: K+0/K+1, 1: K+2/K+3
  int lrow = lane & 15;

  v8f c = {};
  const float* arow = agg + (size_t)(m0 + lrow) * D;
  #pragma unroll
  for (int k = 0; k < D; k += 4) {
    int kb = k + half * 2;
    v2f a, b;
    a.x = arow[kb];
    a.y = arow[kb + 1];
    b.x = kbuf[kb * D + n0 + lrow];
    b.y = kbuf[(kb + 1) * D + n0 + lrow];
    // 8 args: (neg_a, A, neg_b, B, c_mod, C, reuse_a, reuse_b)
    c = __builtin_amdgcn_wmma_f32_16x16x4_f32(false, a, false, b, (short)0, c,
                                              false, false);
  }

  int ncol = n0 + lrow;
  float bv = bias[ncol];
  int rbase = m0 + half * 8;
  #pragma unroll
  for (int r = 0; r < 8; ++r) {
    int row = rbase + r;
    float nrm = rsqrtf(deg[row]);      // deg > 0 guaranteed by self-loops
    float v = c[r] * nrm + bv;
    out[(size_t)row * D + ncol] = v > 0.0f ? v : 0.0f;
  }
}

extern "C" void kernel_launch(void* const* d_in, const int* in_sizes, int n_in,
                              void* d_out, int out_size, void* d_ws, size_t ws_size,
                              hipStream_t stream) {
  const float* h    = (const float*)d_in[0];
  const float* kern = (const float*)d_in[1];
  const float* bias = (const float*)d_in[2];
  const int*   src  = (const int*)d_in[3];
  const int*   dst  = (const int*)d_in[4];
  float* out = (float*)d_out;

  int n_nodes = in_sizes[0] / D;
  int n_edges = in_sizes[3];

  float* agg = (float*)d_ws;                       // n_nodes * 64 floats
  float* deg = agg + (size_t)n_nodes * D;          // n_nodes floats

  // 1) zero accumulators
  size_t zn = (size_t)n_nodes * D + (size_t)n_nodes;
  GCN_zero_ws<<<1024, 256, 0, stream>>>(agg, zn);

  // 2) scatter-sum + degree
  long long sthreads = (long long)n_edges * 16;
  int sblocks = (int)((sthreads + 255) / 256);
  GCN_scatter_deg<<<sblocks, 256, 0, stream>>>(h, src, dst, agg, deg, n_edges);

  // 3) fused WMMA GEMM + norm + bias + relu
  int n_tiles_m = (n_nodes + TILE - 1) / TILE;
  int total_tiles = n_tiles_m * (D / TILE);
  int gblocks = (total_tiles + 7) / 8;
  GCN_gemm_norm_bias_relu<<<gblocks, 256, 0, stream>>>(agg, kern, bias, deg, out,
                                                       n_tiles_m);
}